// GAE_85899345920190
// MI455X (gfx1250) — compile-verified
//
#include <hip/hip_runtime.h>
#include <hip/hip_bf16.h>
#include <math.h>

// GCN autoencoder: 4x (GEMM + normalized scatter-add + bias[/ELU]).
// GEMMs use CDNA5 V_WMMA_F32_16X16X4_F32 (fp32-exact). Scatter is L2-resident
// (all node-feature matrices <= 51MB << 192MB L2), one wave32 per edge.

#define NN      50000
#define NE      800000
#define IN_DIM  256
#define HID_DIM 128
#define OUT_DIM 64

typedef float v2f __attribute__((ext_vector_type(2)));
typedef float v8f __attribute__((ext_vector_type(8)));

// ---------------------------------------------------------------- utilities
__global__ void zero_kernel(float* __restrict__ p, int n) {
    int i = blockIdx.x * blockDim.x + threadIdx.x;
    if (i < n) p[i] = 0.0f;
}

__global__ void deg_init_kernel(float* __restrict__ deg, int n) {
    int i = blockIdx.x * blockDim.x + threadIdx.x;
    if (i < n) deg[i] = 1.0f;          // self loop contributes 1 to every node
}

__global__ void deg_edge_kernel(const long long* __restrict__ col,
                                float* __restrict__ deg, int e) {
    int i = blockIdx.x * blockDim.x + threadIdx.x;
    if (i < e) atomicAdd(&deg[(int)col[i]], 1.0f);
}

__global__ void deg_fin_kernel(float* __restrict__ deg, int n) {
    int i = blockIdx.x * blockDim.x + threadIdx.x;
    if (i < n) {
        float d = deg[i];
        deg[i] = (d > 0.0f) ? rsqrtf(d) : 0.0f;   // becomes dinv
    }
}

// ---------------------------------------------------------------- WMMA GEMM
// C[M,N] = A[M,K] @ B[K,N], row-major, fp32. One wave per 16x16 tile,
// K stepped by 4 with V_WMMA_F32_16X16X4_F32.
// A-frag (16x4 f32): lanes 0-15 -> K={0,1}, lanes 16-31 -> K={2,3}, M = lane&15.
// B-frag (4x16 f32): same K split, N = lane&15.
// C/D (16x16 f32, 8 VGPRs): vgpr r -> M = r (lanes 0-15) / 8+r (lanes 16-31).
__global__ void gemm_f32_wmma(const float* __restrict__ A,
                              const float* __restrict__ B,
                              float* __restrict__ C,
                              int M, int K, int N) {
    const int lane   = threadIdx.x & 31;
    const int waveIB = threadIdx.x >> 5;
    const int wavesB = blockDim.x >> 5;
    const int tile   = blockIdx.x * wavesB + waveIB;

    const int tiles_n = N >> 4;
    const int tiles_m = M >> 4;
    if (tile >= tiles_m * tiles_n) return;     // wave-uniform: EXEC stays full

    const int tm = tile / tiles_n;
    const int tn = tile - tm * tiles_n;
    const int m0 = tm << 4;
    const int n0 = tn << 4;

    const int half = lane >> 4;                // 0: K pair {0,1}, 1: {2,3}
    const int l15  = lane & 15;

    const float* Arow = A + (size_t)(m0 + l15) * K + 2 * half;
    const float* Bcol = B + (size_t)(2 * half) * N + (n0 + l15);

    v8f acc = {};
    for (int k = 0; k < K; k += 4) {
        v2f a, b;
        a.x = Arow[k + 0];
        a.y = Arow[k + 1];
        b.x = Bcol[(size_t)k * N];
        b.y = Bcol[(size_t)(k + 1) * N];
        acc = __builtin_amdgcn_wmma_f32_16x16x4_f32(
            /*neg_a=*/false, a, /*neg_b=*/false, b,
            /*c_mod=*/(short)0, acc, /*reuse_a=*/false, /*reuse_b=*/false);
    }

    float* Crow = C + (size_t)(m0 + 8 * half) * N + n0 + l15;
#pragma unroll
    for (int r = 0; r < 8; ++r)
        Crow[(size_t)r * N] = acc[r];
}

// ---------------------------------------------------------------- scatter
// One wave32 per edge: acc[col] += h[row] * dinv[row]*dinv[col].
// Features striped over lanes -> fully coalesced gather and atomic segments.
__global__ void scatter_kernel(const float* __restrict__ h,
                               const long long* __restrict__ row,
                               const long long* __restrict__ col,
                               const float* __restrict__ dinv,
                               float* __restrict__ acc,
                               int e, int F) {
    const int wave = (blockIdx.x * blockDim.x + threadIdx.x) >> 5;
    const int lane = threadIdx.x & 31;
    if (wave >= e) return;                     // wave-uniform
    const int r = (int)row[wave];
    const int c = (int)col[wave];
    const float norm = dinv[r] * dinv[c];
    const float* hs = h   + (size_t)r * F;
    float*       ad = acc + (size_t)c * F;
    for (int f = lane; f < F; f += 32)
        atomicAdd(&ad[f], hs[f] * norm);
}

// ---------------------------------------------------------------- finalize
// out = acc + h*dinv^2 (self loop) + b, optional ELU. In-place on acc.
__global__ void finalize_kernel(float* __restrict__ out,
                                const float* __restrict__ h,
                                const float* __restrict__ dinv,
                                const float* __restrict__ b,
                                int nn, int F, int do_elu) {
    int idx = blockIdx.x * blockDim.x + threadIdx.x;
    if (idx >= nn * F) return;
    int i = idx / F;
    int f = idx - i * F;
    float d = dinv[i];
    float v = out[idx] + h[idx] * d * d + b[f];
    if (do_elu) v = (v > 0.0f) ? v : (expf(v) - 1.0f);
    out[idx] = v;
}

// ---------------------------------------------------------------- launch
extern "C" void kernel_launch(void* const* d_in, const int* in_sizes, int n_in,
                              void* d_out, int out_size, void* d_ws, size_t ws_size,
                              hipStream_t stream) {
    const float*     features = (const float*)d_in[0];
    const long long* ei       = (const long long*)d_in[1];   // int64 (2, E)
    const float* W1 = (const float*)d_in[2]; const float* b1 = (const float*)d_in[3];
    const float* W2 = (const float*)d_in[4]; const float* b2 = (const float*)d_in[5];
    const float* W3 = (const float*)d_in[6]; const float* b3 = (const float*)d_in[7];
    const float* W4 = (const float*)d_in[8]; const float* b4 = (const float*)d_in[9];

    const long long* row = ei;          // sources
    const long long* col = ei + NE;     // targets

    float* out = (float*)d_out;
    float* h2  = out;                            // [NN, 64]
    float* h4  = out + (size_t)NN * OUT_DIM;     // [NN, 256]

    // Workspace layout (~77 MB): dinv | gemm_buf [NN,256] | h13 [NN,128]
    float* ws   = (float*)d_ws;
    float* dinv = ws;
    float* gbuf = ws + 50048;                    // padded past NN
    float* h13  = gbuf + (size_t)NN * IN_DIM;

    const int T = 256;

    // Degree / normalization (recomputed every call; buffers are poisoned).
    deg_init_kernel<<<(NN + T - 1) / T, T, 0, stream>>>(dinv, NN);
    deg_edge_kernel<<<(NE + T - 1) / T, T, 0, stream>>>(col, dinv, NE);
    deg_fin_kernel <<<(NN + T - 1) / T, T, 0, stream>>>(dinv, NN);

    auto layer = [&](const float* x, const float* W, const float* b,
                     int K, int F, float* acc_out, int elu) {
        const int tiles = (NN / 16) * (F / 16);
        gemm_f32_wmma<<<(tiles + 7) / 8, T, 0, stream>>>(x, W, gbuf, NN, K, F);
        const int tot = NN * F;
        zero_kernel    <<<(tot + T - 1) / T, T, 0, stream>>>(acc_out, tot);
        scatter_kernel <<<(NE + 7) / 8, T, 0, stream>>>(gbuf, row, col, dinv, acc_out, NE, F);
        finalize_kernel<<<(tot + T - 1) / T, T, 0, stream>>>(acc_out, gbuf, dinv, b, NN, F, elu);
    };

    layer(features, W1, b1, IN_DIM,  HID_DIM, h13, 1);   // h1 (ELU) -> h13
    layer(h13,      W2, b2, HID_DIM, OUT_DIM, h2,  0);   // h2       -> d_out[0:NN*64]
    layer(h2,       W3, b3, OUT_DIM, HID_DIM, h13, 1);   // h3 (ELU) -> h13 (reuse)
    layer(h13,      W4, b4, HID_DIM, IN_DIM,  h4,  0);   // h4       -> d_out[NN*64:]
}